// GCN_44435731644594
// MI455X (gfx1250) — compile-verified
//
#include <hip/hip_runtime.h>
#include <math.h>

#define Bc 4
#define WINc 5
#define Nc 50000
#define Ec 800000
#define Hc 32

typedef __attribute__((ext_vector_type(2))) float v2f;
typedef __attribute__((ext_vector_type(8))) float v8f;

// ---------------- utility kernels ----------------

__global__ __launch_bounds__(256) void tag_zero(float* __restrict__ p, long long n) {
    long long i = (long long)blockIdx.x * blockDim.x + threadIdx.x;
    if (i < n) p[i] = 0.0f;
}

__global__ __launch_bounds__(256) void tag_degree(const int* __restrict__ dst,
                                                  float* __restrict__ deg) {
    int e = blockIdx.x * blockDim.x + threadIdx.x;
    if (e < Ec) atomicAdd(&deg[dst[e]], 1.0f);
}

__global__ __launch_bounds__(256) void tag_norm(const int* __restrict__ src,
                                                const int* __restrict__ dst,
                                                const float* __restrict__ deg,
                                                float* __restrict__ norm) {
    int e = blockIdx.x * blockDim.x + threadIdx.x;
    if (e >= Ec) return;
    float a = deg[src[e]], b = deg[dst[e]];
    float ra = (a > 0.0f) ? rsqrtf(a) : 0.0f;
    float rb = (b > 0.0f) ? rsqrtf(b) : 0.0f;
    norm[e] = ra * rb;
}

// X:(B,WIN,N) -> h8:(B*N,8) zero-padded feature-contiguous
__global__ __launch_bounds__(256) void tag_pad_transpose(const float* __restrict__ X,
                                                         float* __restrict__ h8) {
    long long i = (long long)blockIdx.x * blockDim.x + threadIdx.x;
    const long long total = (long long)Bc * Nc * 8;
    if (i >= total) return;
    int f = (int)(i & 7);
    long long bn = i >> 3;
    int n = (int)(bn % Nc);
    int b = (int)(bn / Nc);
    h8[i] = (f < WINc) ? X[((long long)b * WINc + f) * Nc + n] : 0.0f;
}

// propagate: hnext[dst] += norm[e] * hprev[src].
// One wave owns 32 edges of one batch: each lane loads one edge's metadata
// (coalesced), prefetches its gather cacheline (global_prefetch_b8), then the
// wave iterates over the edges broadcasting metadata via shuffles while the
// 32 lanes serve as feature lanes (F contiguous floats per edge row).
template <int F>
__global__ __launch_bounds__(256) void tag_prop(const float* __restrict__ hprev,
                                                float* __restrict__ hnext,
                                                const int* __restrict__ src,
                                                const int* __restrict__ dst,
                                                const float* __restrict__ norm) {
    const long long wavesPerBatch = Ec / 32;            // 25000
    long long w = ((long long)blockIdx.x * blockDim.x + threadIdx.x) >> 5;
    const int lane = threadIdx.x & 31;
    if (w >= (long long)Bc * wavesPerBatch) return;
    const int b = (int)(w / wavesPerBatch);
    const long long e0 = (w % wavesPerBatch) * 32;

    const int   s_l = src[e0 + lane];
    const int   d_l = dst[e0 + lane];
    const float n_l = norm[e0 + lane];

    const float* __restrict__ hs = hprev + (long long)b * Nc * F;
    float* __restrict__       ho = hnext + (long long)b * Nc * F;

    // launch all 32 gather cachelines up front
    __builtin_prefetch(&hs[(long long)s_l * F], 0, 3);

    const int f   = lane & (F - 1);
    const int grp = lane >> ((F == 8) ? 3 : 5);         // edge sub-group within wave
    const int EPI = 32 / F;                              // edges handled per iteration

    for (int j = 0; j < F; ++j) {
        int idx = j * EPI + grp;
        int   s  = __shfl(s_l, idx, 32);
        int   d  = __shfl(d_l, idx, 32);
        float nv = __shfl(n_l, idx, 32);
        float v = nv * hs[(long long)s * F + f];
        atomicAdd(&ho[(long long)d * F + f], v);
    }
}

// ---------------- WMMA GEMM: (M,FPAD) x (FPAD,32) -> (M,32) ----------------
// flags bit0: accumulate from `accum`; bit1: finalize (add bias, tanh)
// One wave computes a 16x32 output tile with V_WMMA_F32_16X16X4_F32.
template <int FPAD>
__global__ __launch_bounds__(256) void tag_gemm(const float* __restrict__ A,
                                                const float* __restrict__ Wk,
                                                int finReal,
                                                const float* __restrict__ accum,
                                                float* __restrict__ dstOut,
                                                const float* __restrict__ bias,
                                                int flags) {
    __shared__ float wlds[FPAD * Hc];
    for (int i = threadIdx.x; i < FPAD * Hc; i += 256) {
        int f = i >> 5;
        wlds[i] = (f < finReal) ? Wk[f * Hc + (i & 31)] : 0.0f;  // zero-pad K
    }
    __syncthreads();

    const int lane = threadIdx.x & 31;
    const int wv = threadIdx.x >> 5;
    const int tile = blockIdx.x * 8 + wv;           // 16 rows per tile, wave-uniform
    const int Mtiles = (Bc * Nc) / 16;              // 12500 (exact)
    if (tile >= Mtiles) return;                     // uniform branch: EXEC all-1 at WMMA

    const int m0 = tile * 16;
    const int r = lane & 15;                        // row within tile / column within N-tile
    const int kh = (lane >> 4) << 1;                // K offset: 0 (lanes 0-15) or 2 (16-31)

    v8f c0 = {};                                    // N columns 0..15
    v8f c1 = {};                                    // N columns 16..31
    const float* arow = A + (long long)(m0 + r) * FPAD + kh;

#pragma unroll
    for (int kk = 0; kk < FPAD / 4; ++kk) {
        v2f a;                                      // A 16x4 f32 fragment
        a.x = arow[kk * 4 + 0];
        a.y = arow[kk * 4 + 1];
        const float* wp = &wlds[(kk * 4 + kh) * Hc];
        v2f bA, bB;                                 // B 4x16 f32 fragments
        bA.x = wp[r];          bA.y = wp[Hc + r];
        bB.x = wp[16 + r];     bB.y = wp[Hc + 16 + r];
        c0 = __builtin_amdgcn_wmma_f32_16x16x4_f32(false, a, false, bA, (short)0, c0, false, false);
        c1 = __builtin_amdgcn_wmma_f32_16x16x4_f32(false, a, false, bB, (short)0, c1, false, false);
    }

    const int mb = m0 + ((lane >> 4) << 3);         // C/D layout: lanes 16-31 hold rows M+8
    float bs0 = 0.0f, bs1 = 0.0f;
    if (flags & 2) { bs0 = bias[r]; bs1 = bias[16 + r]; }
#pragma unroll
    for (int j = 0; j < 8; ++j) {
        long long o0 = (long long)(mb + j) * Hc + r;
        long long o1 = o0 + 16;
        float v0 = c0[j], v1 = c1[j];
        if (flags & 1) { v0 += accum[o0]; v1 += accum[o1]; }
        if (flags & 2) { v0 = tanhf(v0 + bs0); v1 = tanhf(v1 + bs1); }
        dstOut[o0] = v0;
        dstOut[o1] = v1;
    }
}

// head: pred[i] = sum_f h[i,f]*Wr[f] + br ; one wave per row, shuffle reduce
__global__ __launch_bounds__(256) void tag_head(const float* __restrict__ h,
                                                const float* __restrict__ Wr,
                                                const float* __restrict__ br,
                                                float* __restrict__ pred) {
    int gw = (int)(((long long)blockIdx.x * blockDim.x + threadIdx.x) >> 5);
    int lane = threadIdx.x & 31;
    if (gw >= Bc * Nc) return;
    float v = h[(long long)gw * Hc + lane] * Wr[lane];
#pragma unroll
    for (int off = 16; off > 0; off >>= 1) v += __shfl_xor(v, off, 32);
    if (lane == 0) pred[gw] = v + br[0];
}

// ---------------- host-side orchestration ----------------

static inline void launch_zero(float* p, long long n, hipStream_t s) {
    tag_zero<<<(int)((n + 255) / 256), 256, 0, s>>>(p, n);
}

template <int FPAD>
static void tag_layer(float* bufA, float* bufB, const float* W, const float* bias,
                      const int* src, const int* dst, const float* norm,
                      float* out, float* hdst, int finReal, hipStream_t s) {
    const int GB = ((Bc * Nc / 16) + 7) / 8;                 // 1563 blocks, 8 waves each
    const int PB = (Bc * (Ec / 32) * 32 + 255) / 256;        // one wave per 32 edges
    const long long bufN = (long long)Bc * Nc * FPAD;
    const int ws = finReal * Hc;                             // stride between hop matrices

    tag_gemm<FPAD><<<GB, 256, 0, s>>>(bufA, W, finReal, out, out, nullptr, 0);
    launch_zero(bufB, bufN, s);
    tag_prop<FPAD><<<PB, 256, 0, s>>>(bufA, bufB, src, dst, norm);
    tag_gemm<FPAD><<<GB, 256, 0, s>>>(bufB, W + 1 * ws, finReal, out, out, nullptr, 1);
    launch_zero(bufA, bufN, s);
    tag_prop<FPAD><<<PB, 256, 0, s>>>(bufB, bufA, src, dst, norm);
    tag_gemm<FPAD><<<GB, 256, 0, s>>>(bufA, W + 2 * ws, finReal, out, out, nullptr, 1);
    launch_zero(bufB, bufN, s);
    tag_prop<FPAD><<<PB, 256, 0, s>>>(bufA, bufB, src, dst, norm);
    tag_gemm<FPAD><<<GB, 256, 0, s>>>(bufB, W + 3 * ws, finReal, out, hdst, bias, 3);
}

extern "C" void kernel_launch(void* const* d_in, const int* in_sizes, int n_in,
                              void* d_out, int out_size, void* d_ws, size_t ws_size,
                              hipStream_t stream) {
    (void)in_sizes; (void)n_in; (void)out_size; (void)ws_size;

    const float* X  = (const float*)d_in[0];
    const int*   ei = (const int*)d_in[1];
    const float* W0 = (const float*)d_in[2];
    const float* b0 = (const float*)d_in[3];
    const float* W1 = (const float*)d_in[4];
    const float* b1 = (const float*)d_in[5];
    const float* W2 = (const float*)d_in[6];
    const float* b2 = (const float*)d_in[7];
    const float* Wr = (const float*)d_in[8];
    const float* br = (const float*)d_in[9];
    const int* src = ei;
    const int* dst = ei + Ec;

    float* ws  = (float*)d_ws;
    float* deg  = ws;  ws += Nc;
    float* norm = ws;  ws += Ec;
    float* h8a  = ws;  ws += (long long)Bc * Nc * 8;
    float* h8b  = ws;  ws += (long long)Bc * Nc * 8;
    float* ha   = ws;  ws += (long long)Bc * Nc * 32;
    float* hb   = ws;  ws += (long long)Bc * Nc * 32;
    float* outb = ws;  ws += (long long)Bc * Nc * 32;

    float* pred = (float*)d_out;                       // (B,1,N) flat = b*N+n
    float* hfin = (float*)d_out + (long long)Bc * Nc;  // (B*N, 32)

    // gcn_norm
    launch_zero(deg, Nc, stream);
    tag_degree<<<(Ec + 255) / 256, 256, 0, stream>>>(dst, deg);
    tag_norm<<<(Ec + 255) / 256, 256, 0, stream>>>(src, dst, deg, norm);

    // transpose + pad input to (B*N, 8)
    {
        long long t = (long long)Bc * Nc * 8;
        tag_pad_transpose<<<(int)((t + 255) / 256), 256, 0, stream>>>(X, h8a);
    }

    // layer 0: 5->32 (K padded to 8), layers 1,2: 32->32
    tag_layer<8>(h8a, h8b, W0, b0, src, dst, norm, outb, ha, WINc, stream);
    tag_layer<32>(ha, hb, W1, b1, src, dst, norm, outb, ha, Hc, stream);
    tag_layer<32>(ha, hb, W2, b2, src, dst, norm, outb, hfin, Hc, stream);

    // prediction head
    tag_head<<<(Bc * Nc * 32 + 255) / 256, 256, 0, stream>>>(hfin, Wr, br, pred);
}